// Model_50328426774833
// MI455X (gfx1250) — compile-verified
//
#include <hip/hip_runtime.h>
#include <hip/hip_bf16.h>

typedef __attribute__((ext_vector_type(2))) float v2f;
typedef __attribute__((ext_vector_type(8))) float v8f;

#define DIM 64
#define OUTW 256   // 4*DIM output width

// ---------------------------------------------------------------------------
// Copy x [N,64] into d_out[:, 0:64] (row stride 256 floats), float4 granules.
// ---------------------------------------------------------------------------
__global__ void kgat_copy_x(const float4* __restrict__ x4,
                            float4* __restrict__ out4, int n) {
    int i = blockIdx.x * blockDim.x + threadIdx.x;   // over n*16 float4s
    if (i >= n * 16) return;
    int row = i >> 4;
    int c4  = i & 15;
    out4[row * (OUTW / 4) + c4] = x4[i];
}

// ---------------------------------------------------------------------------
// Zero h_n (float4 granules).
// ---------------------------------------------------------------------------
__global__ void kgat_zero(float4* __restrict__ p, int n4) {
    int i = blockIdx.x * blockDim.x + threadIdx.x;
    if (i >= n4) return;
    p[i] = make_float4(0.f, 0.f, 0.f, 0.f);
}

// ---------------------------------------------------------------------------
// Edge kernel: one wave32 per edge. Lane l handles features 2l, 2l+1.
// msg = h[src] * a[e]; atomicAdd into h_n[dst].
// h lives in d_out at column offset in_col (row stride 256).
// ---------------------------------------------------------------------------
__global__ void __launch_bounds__(256)
kgat_edge(const float* __restrict__ hbase, int in_col,
          const float* __restrict__ a,
          const int* __restrict__ src, const int* __restrict__ dst,
          float* __restrict__ hn, int n_edges) {
    int wave = (blockIdx.x * blockDim.x + threadIdx.x) >> 5;
    int lane = threadIdx.x & 31;
    if (wave >= n_edges) return;

    int s   = src[wave];
    int d   = dst[wave];
    float w = a[wave];

    const float2 hv = *(const float2*)(hbase + (size_t)s * OUTW + in_col + lane * 2);
    float* p = hn + (size_t)d * DIM + lane * 2;
    atomicAdd(p + 0, hv.x * w);
    atomicAdd(p + 1, hv.y * w);
}

// ---------------------------------------------------------------------------
// Node kernel: h_out = LeakyReLU((h+h_n)@W1 + b1) + LeakyReLU((h*h_n)@W2 + b2)
// 256 threads = 8 waves; each wave computes a 16x64 output tile with
// v_wmma_f32_16x16x4_f32 (16 K-steps of K=4 over DIM=64, 4 N-tiles, 2 mats).
//
// LDS weight layout is PAIR-INTERLEAVED so each lane's B fragment
// {W[kb][col], W[kb+1][col]} (kb even) is one aligned 8-byte ds_load_b64:
//   element (r, c)  ->  sW[(r>>1)*128 + c*2 + (r&1)]
// ---------------------------------------------------------------------------
__global__ void __launch_bounds__(256)
kgat_node(const float* __restrict__ hbase,   // d_out (read at in_col)
          const float* __restrict__ hn,      // [N,64]
          const float* __restrict__ W1, const float* __restrict__ b1,
          const float* __restrict__ W2, const float* __restrict__ b2,
          float* __restrict__ out,           // d_out (write at out_col)
          int n, int in_col, int out_col) {
    __shared__ float sW1[DIM * DIM];
    __shared__ float sW2[DIM * DIM];

    // Cooperative weight stage with pair-interleave transpose.
    {
        const float4* w1v = (const float4*)W1;
        const float4* w2v = (const float4*)W2;
        for (int i = threadIdx.x; i < (DIM * DIM / 4); i += 256) {
            const int r  = i >> 4;           // row 0..63
            const int c0 = (i & 15) * 4;     // column base
            const float4 w1 = w1v[i];
            const float4 w2 = w2v[i];
            const int base = (r >> 1) * 128 + (r & 1);
            sW1[base + (c0 + 0) * 2] = w1.x;
            sW1[base + (c0 + 1) * 2] = w1.y;
            sW1[base + (c0 + 2) * 2] = w1.z;
            sW1[base + (c0 + 3) * 2] = w1.w;
            sW2[base + (c0 + 0) * 2] = w2.x;
            sW2[base + (c0 + 1) * 2] = w2.y;
            sW2[base + (c0 + 2) * 2] = w2.z;
            sW2[base + (c0 + 3) * 2] = w2.w;
        }
    }
    __syncthreads();

    const int waveId = threadIdx.x >> 5;
    const int lane   = threadIdx.x & 31;
    const int half   = lane >> 4;     // 0: K=0,1 ; 1: K=2,3  (f32 A/B layout)
    const int m      = lane & 15;

    const int row0 = blockIdx.x * 128 + waveId * 16;
    int rowA = row0 + m;
    if (rowA >= n) rowA = n - 1;      // clamp; WMMA needs full EXEC, stores guarded

    const float* hrow = hbase + (size_t)rowA * OUTW + in_col + 2 * half;
    const float* nrow = hn + (size_t)rowA * DIM + 2 * half;

    v8f acc1[4] = {};   // (h+h_n)@W1 accumulators, 4 N-tiles
    v8f acc2[4] = {};   // (h*h_n)@W2 accumulators

#pragma unroll
    for (int ks = 0; ks < 16; ++ks) {
        const int k = ks * 4;
        // A fragments: lane holds A[m][k+2*half], A[m][k+2*half+1]
        const float2 hv = *(const float2*)(hrow + k);
        const float2 nv = *(const float2*)(nrow + k);
        v2f aSum = { hv.x + nv.x, hv.y + nv.y };
        v2f aPrd = { hv.x * nv.x, hv.y * nv.y };

        // B pair rows: kb = k + 2*half (even), pair index j = 2*ks + half
        const int jbase = (2 * ks + half) * 128;
#pragma unroll
        for (int nt = 0; nt < 4; ++nt) {
            const int col = nt * 16 + m;
            const v2f bw1 = *(const v2f*)&sW1[jbase + col * 2];
            const v2f bw2 = *(const v2f*)&sW2[jbase + col * 2];
            acc1[nt] = __builtin_amdgcn_wmma_f32_16x16x4_f32(
                false, aSum, false, bw1, (short)0, acc1[nt], false, false);
            acc2[nt] = __builtin_amdgcn_wmma_f32_16x16x4_f32(
                false, aPrd, false, bw2, (short)0, acc2[nt], false, false);
        }
    }

    // Epilogue: C/D layout — VGPR r, lane: M = r + 8*half, N = nt*16 + m.
#pragma unroll
    for (int nt = 0; nt < 4; ++nt) {
        const int col = nt * 16 + m;
        const float bb1 = b1[col];
        const float bb2 = b2[col];
#pragma unroll
        for (int r = 0; r < 8; ++r) {
            const int rowD = row0 + r + 8 * half;
            if (rowD < n) {
                float v1 = acc1[nt][r] + bb1;
                float v2 = acc2[nt][r] + bb2;
                v1 = v1 > 0.f ? v1 : 0.01f * v1;
                v2 = v2 > 0.f ? v2 : 0.01f * v2;
                out[(size_t)rowD * OUTW + out_col + col] = v1 + v2;
            }
        }
    }
}

// ---------------------------------------------------------------------------
// Launch: copy x -> out[:,0:64]; per layer: zero hn, edge scatter, node GEMM.
// ---------------------------------------------------------------------------
extern "C" void kernel_launch(void* const* d_in, const int* in_sizes, int n_in,
                              void* d_out, int out_size, void* d_ws, size_t ws_size,
                              hipStream_t stream) {
    const float* x   = (const float*)d_in[0];
    const float* a   = (const float*)d_in[1];
    const float* W1s = (const float*)d_in[2];
    const float* b1s = (const float*)d_in[3];
    const float* W2s = (const float*)d_in[4];
    const float* b2s = (const float*)d_in[5];
    const int*   src = (const int*)d_in[6];
    const int*   dst = (const int*)d_in[7];
    float* out = (float*)d_out;
    float* hn  = (float*)d_ws;             // [N, 64] scratch

    const int n       = in_sizes[0] / DIM; // 100000
    const int n_edges = in_sizes[1];       // 3200000

    // x -> out[:, 0:64]
    {
        int work = n * 16;  // float4s
        kgat_copy_x<<<(work + 255) / 256, 256, 0, stream>>>(
            (const float4*)x, (float4*)out, n);
    }

    for (int l = 0; l < 3; ++l) {
        const int in_col  = l * DIM;
        const int out_col = (l + 1) * DIM;

        int n4 = n * (DIM / 4);
        kgat_zero<<<(n4 + 255) / 256, 256, 0, stream>>>((float4*)hn, n4);

        int edge_blocks = (n_edges + 7) / 8;   // 8 waves/block, 1 edge/wave
        kgat_edge<<<edge_blocks, 256, 0, stream>>>(
            out, in_col, a, src, dst, hn, n_edges);

        int node_blocks = (n + 127) / 128;     // 8 waves/block, 16 rows/wave
        kgat_node<<<node_blocks, 256, 0, stream>>>(
            out, hn,
            W1s + (size_t)l * DIM * DIM, b1s + (size_t)l * DIM,
            W2s + (size_t)l * DIM * DIM, b2s + (size_t)l * DIM,
            out, n, in_col, out_col);
    }
}